// MatrixAttentionFunc_21869973471316
// MI455X (gfx1250) — compile-verified
//
#include <hip/hip_runtime.h>
#include <hip/hip_bf16.h>

typedef __attribute__((ext_vector_type(2))) float v2f;
typedef __attribute__((ext_vector_type(8))) float v8f;
typedef __attribute__((ext_vector_type(4))) unsigned int u32x4;
typedef __attribute__((ext_vector_type(8))) int i32x8;
typedef __attribute__((ext_vector_type(4))) int i32x4;

#define NH   8
#define KD   32
#define HD   32
#define DCH  256      // NH*HD
#define OC   1280     // NH*(4*KD+HD)
#define SPAT 64
#define HW   4096     // 64*64
#define ATTN_SCALE 0.17677669529663687f

#if defined(__has_builtin)
#if __has_builtin(__builtin_amdgcn_tensor_load_to_lds) && __has_builtin(__builtin_amdgcn_s_wait_tensorcnt)
#define HAVE_TDM 1
#endif
#endif

__device__ __forceinline__ v8f wmma_f32(v2f a, v2f b, v8f c) {
    // D = A(16x4,f32) * B(4x16,f32) + C(16x16,f32)
    return __builtin_amdgcn_wmma_f32_16x16x4_f32(
        /*neg_a=*/false, a, /*neg_b=*/false, b,
        /*c_mod=*/(short)0, c, /*reuse_a=*/false, /*reuse_b=*/false);
}

#ifdef HAVE_TDM
// TDM: DMA a 2D tile (tile_dim0=64 elems/row, tile_dim1=32 rows, row stride
// 4096 elems, f32) from global into LDS at byte offset lds_off.
// D# packing per CDNA5 ISA section 8.3 (group 0) / 8.4 (group 1).
// This toolchain exposes the 6-arg builtin: (g0, g1, g2, g3, g_extra, cpol).
__device__ __forceinline__ void tdm_load_panel(const float* gsrc, unsigned lds_off) {
    const unsigned long long ga = (unsigned long long)(uintptr_t)gsrc;
    u32x4 g0;
    g0.x = 1u;                                                 // count=1 (valid user D#)
    g0.y = lds_off;                                            // lds_addr (bytes)
    g0.z = (unsigned)(ga & 0xFFFFFFFFu);                       // global_addr[31:0]
    g0.w = (unsigned)((ga >> 32) & 0x1FFFFFFu) | 0x80000000u;  // global_addr[56:32] | type=2
    i32x8 g1;
    g1[0] = 0x20000;        // workgroup_mask=0, data_size=2 (4 bytes), no flags
    g1[1] = 64 << 16;       // tensor_dim0 = 64 (lo16 in [31:16])
    g1[2] = 32 << 16;       // tensor_dim0 hi=0 | tensor_dim1 = 32
    g1[3] = 64 << 16;       // tensor_dim1 hi=0 | tile_dim0 = 64
    g1[4] = 32;             // tile_dim1 = 32 | tile_dim2 = 0
    g1[5] = 4096;           // tensor_dim0_stride = 4096 elements
    g1[6] = 0;              // stride0 hi | stride1 lo
    g1[7] = 0;              // stride1 hi
    const i32x4 z4 = {0, 0, 0, 0};                             // groups 2/3 unused (2D)
    const i32x8 z8 = {0, 0, 0, 0, 0, 0, 0, 0};
    __builtin_amdgcn_tensor_load_to_lds(g0, g1, z4, z4, z8, 0);
}
#endif

// -------------------------------------------------------------------------
// Stage 1: projection  P[b, o, hw] = sum_c W[o,c] * X[b, c, hw]
// One 16x16 output tile per wave, K=256 in steps of 4 (fp32 WMMA).
// -------------------------------------------------------------------------
__global__ __launch_bounds__(256)
void proj_kernel(const float* __restrict__ x, const float* __restrict__ w,
                 float* __restrict__ P) {
    const int lane = threadIdx.x & 31;
    const int wv   = threadIdx.x >> 5;
    const int gw   = blockIdx.x * 8 + wv;          // global wave id
    const int b    = gw / (80 * 256);
    const int rm   = gw % (80 * 256);
    const int o0   = (rm / 256) * 16;              // 1280/16 = 80 tiles
    const int n0   = (rm % 256) * 16;              // 4096/16 = 256 tiles
    const int m    = lane & 15;                    // row / col within tile
    const int kh   = lane >> 4;                    // K half-select

    const float* wrow = w + (size_t)(o0 + m) * DCH;
    const float* xcol = x + (size_t)b * DCH * HW + n0 + m;

    v8f acc = {};
    for (int k0 = 0; k0 < DCH; k0 += 4) {
        const int kk = k0 + 2 * kh;
        v2f a, bb;
        a.x  = wrow[kk];
        a.y  = wrow[kk + 1];
        bb.x = xcol[(size_t)kk * HW];
        bb.y = xcol[(size_t)(kk + 1) * HW];
        acc = wmma_f32(a, bb, acc);
    }
    float* pout = P + ((size_t)b * OC + o0 + 8 * kh) * HW + n0 + m;
#pragma unroll
    for (int r = 0; r < 8; ++r) pout[(size_t)r * HW] = acc[r];
}

// -------------------------------------------------------------------------
// Stage 2: row attention.  Per (b,H,w): S[i,j] = SCALE * sum_d rq[d,i,w]*rk[d,j,w]
// softmax over i (per column j); store R[b,H,i,j,w].
// -------------------------------------------------------------------------
__global__ __launch_bounds__(256)
void row_attn_kernel(const float* __restrict__ P, float* __restrict__ R) {
    __shared__ float S[64 * 64];
    const int lane = threadIdx.x & 31;
    const int wv   = threadIdx.x >> 5;
    const int b    = blockIdx.x / (NH * SPAT);
    const int rm   = blockIdx.x % (NH * SPAT);
    const int H    = rm / SPAT;
    const int w    = rm % SPAT;
    const int m    = lane & 15;
    const int kh   = lane >> 4;

    const float* rq = P + ((size_t)b * OC + H * 160) * HW;        // 0*KD
    const float* rk = rq + (size_t)KD * HW;                       // 1*KD

    for (int tt = 0; tt < 2; ++tt) {                 // 16 tiles / 8 waves
        const int t  = wv * 2 + tt;
        const int i0 = (t >> 2) * 16;
        const int j0 = (t & 3) * 16;
        v8f acc = {};
        for (int k0 = 0; k0 < KD; k0 += 4) {
            const int kk = k0 + 2 * kh;
            v2f a, bb;
            a.x  = rq[(size_t)kk * HW + (i0 + m) * SPAT + w];
            a.y  = rq[(size_t)(kk + 1) * HW + (i0 + m) * SPAT + w];
            bb.x = rk[(size_t)kk * HW + (j0 + m) * SPAT + w];
            bb.y = rk[(size_t)(kk + 1) * HW + (j0 + m) * SPAT + w];
            acc = wmma_f32(a, bb, acc);
        }
#pragma unroll
        for (int r = 0; r < 8; ++r)
            S[(i0 + r + 8 * kh) * 64 + j0 + m] = acc[r] * ATTN_SCALE;
    }
    __syncthreads();

    if (threadIdx.x < 64) {                           // column-wise softmax over i
        const int j = threadIdx.x;
        float mx = -1e30f;
        for (int i = 0; i < 64; ++i) mx = fmaxf(mx, S[i * 64 + j]);
        float sum = 0.0f;
        for (int i = 0; i < 64; ++i) {
            float e = __expf(S[i * 64 + j] - mx);
            S[i * 64 + j] = e;
            sum += e;
        }
        const float inv = 1.0f / sum;
        float* rout = R + (size_t)(b * NH + H) * 64 * HW + j * SPAT + w;
        for (int i = 0; i < 64; ++i) rout[(size_t)i * HW] = S[i * 64 + j] * inv;
    }
}

// -------------------------------------------------------------------------
// Stage 3: column attention.  Per (b,H,h): T[a,w2] = SCALE*sum_d cq[d,h,a]*ck[d,h,w2]
// softmax over a; store Cm[b,H,a,h,w2]  (layout matching its use in stage 4).
// -------------------------------------------------------------------------
__global__ __launch_bounds__(256)
void col_attn_kernel(const float* __restrict__ P, float* __restrict__ Cm) {
    __shared__ float S[64 * 64];
    const int lane = threadIdx.x & 31;
    const int wv   = threadIdx.x >> 5;
    const int b    = blockIdx.x / (NH * SPAT);
    const int rm   = blockIdx.x % (NH * SPAT);
    const int H    = rm / SPAT;
    const int h    = rm % SPAT;
    const int m    = lane & 15;
    const int kh   = lane >> 4;

    const float* cq = P + ((size_t)b * OC + H * 160 + 2 * KD) * HW;
    const float* ck = P + ((size_t)b * OC + H * 160 + 3 * KD) * HW;

    for (int tt = 0; tt < 2; ++tt) {
        const int t  = wv * 2 + tt;
        const int a0 = (t >> 2) * 16;
        const int w0 = (t & 3) * 16;
        v8f acc = {};
        for (int k0 = 0; k0 < KD; k0 += 4) {
            const int kk = k0 + 2 * kh;
            v2f a, bb;
            a.x  = cq[(size_t)kk * HW + h * SPAT + a0 + m];
            a.y  = cq[(size_t)(kk + 1) * HW + h * SPAT + a0 + m];
            bb.x = ck[(size_t)kk * HW + h * SPAT + w0 + m];
            bb.y = ck[(size_t)(kk + 1) * HW + h * SPAT + w0 + m];
            acc = wmma_f32(a, bb, acc);
        }
#pragma unroll
        for (int r = 0; r < 8; ++r)
            S[(a0 + r + 8 * kh) * 64 + w0 + m] = acc[r] * ATTN_SCALE;
    }
    __syncthreads();

    if (threadIdx.x < 64) {
        const int w2 = threadIdx.x;
        float mx = -1e30f;
        for (int a = 0; a < 64; ++a) mx = fmaxf(mx, S[a * 64 + w2]);
        float sum = 0.0f;
        for (int a = 0; a < 64; ++a) {
            float e = __expf(S[a * 64 + w2] - mx);
            S[a * 64 + w2] = e;
            sum += e;
        }
        const float inv = 1.0f / sum;
        float* cout = Cm + (size_t)(b * NH + H) * 64 * HW + h * SPAT + w2;
        for (int a = 0; a < 64; ++a) cout[(size_t)a * HW] = S[a * 64 + w2] * inv;
    }
}

// -------------------------------------------------------------------------
// Stage 4: fused m + output.  Per (b,H,w) workgroup:
//   out[d,h] = sum_i Rw[i,h] * (Vi[d,:] x Cw[:,h])  + v[d,h,w]
// V panels double-buffered in LDS via the Tensor Data Mover: wave 0 issues
// TENSOR_LOAD_TO_LDS for panel i+1 while all 8 waves run WMMAs on panel i.
// -------------------------------------------------------------------------
__global__ __launch_bounds__(256)
void out_kernel(const float* __restrict__ P, const float* __restrict__ R,
                const float* __restrict__ Cm, float* __restrict__ out) {
    __shared__ float Rw[64 * 64];
    __shared__ float Cw[64 * 64];
    __shared__ float Vi[2][32 * 64];
    const int tid  = threadIdx.x;
    const int lane = tid & 31;
    const int wv   = tid >> 5;
    const int b    = blockIdx.x / (NH * SPAT);
    const int rm   = blockIdx.x % (NH * SPAT);
    const int H    = rm / SPAT;
    const int w    = rm % SPAT;

    const float* rbase = R  + (size_t)(b * NH + H) * 64 * HW;
    const float* cbase = Cm + (size_t)(b * NH + H) * 64 * HW;
    for (int t = tid; t < 4096; t += 256) {
        const int i = t >> 6, h = t & 63;
        Rw[t] = rbase[(size_t)i * HW + h * SPAT + w];
        Cw[t] = cbase[(size_t)i * HW + h * SPAT + w];
    }

    const float* vbase = P + ((size_t)b * OC + H * 160 + 4 * KD) * HW;
    const int d0 = (wv & 1) * 16;          // 2 d-tiles x 4 h-tiles = 8 waves
    const int h0 = (wv >> 1) * 16;
    const int m  = lane & 15;
    const int kh = lane >> 4;

#ifdef HAVE_TDM
    const unsigned vi_off0 = (unsigned)(uintptr_t)(void*)&Vi[0][0];
    const unsigned vi_off1 = (unsigned)(uintptr_t)(void*)&Vi[1][0];
    if (wv == 0) tdm_load_panel(vbase, vi_off0);           // panel 0 in flight
#endif

    v8f acc = {};
    for (int i = 0; i < SPAT; ++i) {
        const int cur = i & 1;
#ifdef HAVE_TDM
        if (wv == 0) __builtin_amdgcn_s_wait_tensorcnt(0); // panel i landed
        __syncthreads();                                   // release + reuse guard
        if (wv == 0 && i + 1 < SPAT)                       // DMA panel i+1 now,
            tdm_load_panel(vbase + (i + 1) * SPAT,         // overlaps the WMMAs
                           cur ? vi_off0 : vi_off1);
#else
        __syncthreads();                                   // reuse guard
        for (int t = tid; t < 2048; t += 256) {
            const int d = t >> 6, j = t & 63;
            Vi[cur][t] = vbase[(size_t)d * HW + i * SPAT + j];
        }
        if (i + 1 < SPAT)
            __builtin_prefetch(&vbase[(size_t)(tid & 31) * HW + (i + 1) * SPAT], 0, 1);
        __syncthreads();
#endif
        const float* V = Vi[cur];
        v8f mi = {};
#pragma unroll
        for (int k0 = 0; k0 < SPAT; k0 += 4) {             // K=64 over j
            const int kk = k0 + 2 * kh;
            v2f a, bb;
            a.x  = V[(d0 + m) * 64 + kk];
            a.y  = V[(d0 + m) * 64 + kk + 1];
            bb.x = Cw[kk * 64 + h0 + m];
            bb.y = Cw[(kk + 1) * 64 + h0 + m];
            mi = wmma_f32(a, bb, mi);
        }
        const float rv = Rw[i * 64 + h0 + m];
#pragma unroll
        for (int r = 0; r < 8; ++r) acc[r] += rv * mi[r];
    }

    float* obase = out + ((size_t)b * DCH + H * HD) * HW;
#pragma unroll
    for (int r = 0; r < 8; ++r) {
        const int d = d0 + r + 8 * kh;
        const int h = h0 + m;
        obase[(size_t)d * HW + h * SPAT + w] =
            acc[r] + vbase[(size_t)d * HW + h * SPAT + w];
    }
}

// -------------------------------------------------------------------------
extern "C" void kernel_launch(void* const* d_in, const int* in_sizes, int n_in,
                              void* d_out, int out_size, void* d_ws, size_t ws_size,
                              hipStream_t stream) {
    (void)in_sizes; (void)n_in; (void)out_size; (void)ws_size;
    const float* x = (const float*)d_in[0];     // (2,256,64,64)
    const float* w = (const float*)d_in[1];     // (1280,256)
    float* outp = (float*)d_out;                // (2,256,64,64)

    float* P  = (float*)d_ws;                   // 2*1280*4096 floats
    float* Rr = P  + (size_t)2 * OC * HW;       // 2*8*64*4096 floats
    float* Cc = Rr + (size_t)2 * NH * 64 * HW;  // 2*8*64*4096 floats

    proj_kernel    <<<5120, 256, 0, stream>>>(x, w, P);
    row_attn_kernel<<<1024, 256, 0, stream>>>(P, Rr);
    col_attn_kernel<<<1024, 256, 0, stream>>>(P, Cc);
    out_kernel     <<<1024, 256, 0, stream>>>(P, Rr, Cc, outp);
}